// QuantumSelfAttention_65481071407021
// MI455X (gfx1250) — compile-verified
//
#include <hip/hip_runtime.h>
#include <math.h>

typedef __attribute__((ext_vector_type(2))) float v2f;
typedef __attribute__((ext_vector_type(8))) float v8f;

#define WIRES 12
#define BATCH_PER_BLOCK 16
#define ROWS_PER_BLOCK (BATCH_PER_BLOCK * WIRES)   // 192 flattened (b,t) rows
#define TILES_PER_BLOCK (ROWS_PER_BLOCK / 16)      // 12 M-tiles of 16 rows
#define THREADS 128                                 // 4 waves (wave32)
#define WAVES (THREADS / 32)
#define TILES_PER_WAVE (TILES_PER_BLOCK / WAVES)    // 3

// out[(b*12+t)*12 + n] = sum_e q[b,t,e] * Wout[n*12+e] + bout[n]
// q via closed-form cosine prefix products (see analysis).
__global__ __launch_bounds__(THREADS)
void qsa_kernel(const float* __restrict__ x,      // (B, 12, 12) [b][w][e]
                const float* __restrict__ theta,  // (12)
                const float* __restrict__ Wout,   // (12, 12) [n][e]
                const float* __restrict__ bout,   // (12)
                float* __restrict__ out)          // (B, 12, 12) [b][t][n]
{
    __shared__ float qs[ROWS_PER_BLOCK][WIRES];   // A-tile staging: [local (b,t) row][e]
    __shared__ float th[WIRES];

    const int tid = threadIdx.x;
    const int b0  = blockIdx.x * BATCH_PER_BLOCK;

    if (tid < WIRES) th[tid] = theta[tid];
    __syncthreads();

    // ---- Stage 1: per (batch, head e): 12 cosines + prefix products -> qs ----
    for (int p = tid; p < BATCH_PER_BLOCK * WIRES; p += THREADS) {
        const int bb = p / WIRES;   // local batch
        const int e  = p % WIRES;   // head index
        const float* xp = x + (size_t)(b0 + bb) * (WIRES * WIRES) + e;

        float c[WIRES];
        #pragma unroll
        for (int w = 0; w < WIRES; ++w)
            c[w] = cosf(xp[w * WIRES] + th[w]);

        // q[t>=1] = prod_{w<=t} c_w ; q[0] = prod_{w=1..11} c_w
        float pre = c[0];
        float q1toN = 1.0f;
        #pragma unroll
        for (int w = 1; w < WIRES; ++w) q1toN *= c[w];
        qs[bb * WIRES + 0][e] = q1toN;
        #pragma unroll
        for (int t = 1; t < WIRES; ++t) {
            pre *= c[t];
            qs[bb * WIRES + t][e] = pre;
        }
    }
    __syncthreads();

    // ---- Stage 2: GEMM (192 x 12) @ (12 x 12) via V_WMMA_F32_16X16X4_F32 ----
    const int lane = tid & 31;
    const int half = lane >> 4;     // 0: lanes 0-15, 1: lanes 16-31
    const int m    = lane & 15;     // A row (lower 4 bits) / B,C,D column n

    // B fragments for the 3 K-steps (kb = e-base 0,4,8).
    // B 4x16 layout: VGPR0 = rows K={0 | 2}, VGPR1 = rows K={1 | 3} (upper lanes -> +2).
    // Pad columns n=12..15 with zeros (predicated loads only; EXEC full at WMMA).
    v2f bf[3];
    #pragma unroll
    for (int kb = 0; kb < 3; ++kb) {
        const int e0 = kb * 4 + 2 * half;
        bf[kb].x = (m < WIRES) ? Wout[m * WIRES + e0]     : 0.0f;
        bf[kb].y = (m < WIRES) ? Wout[m * WIRES + e0 + 1] : 0.0f;
    }
    const float bias = (m < WIRES) ? bout[m] : 0.0f;

    const int wave = tid >> 5;
    for (int t = wave * TILES_PER_WAVE; t < (wave + 1) * TILES_PER_WAVE; ++t) {
        v8f acc = {};
        #pragma unroll
        for (int kb = 0; kb < 3; ++kb) {
            // A 16x4 layout: lane m holds K={0,1}, lane m+16 holds K={2,3}
            const int e0 = kb * 4 + 2 * half;
            v2f a;
            a.x = qs[t * 16 + m][e0];
            a.y = qs[t * 16 + m][e0 + 1];
            acc = __builtin_amdgcn_wmma_f32_16x16x4_f32(
                /*neg_a=*/false, a, /*neg_b=*/false, bf[kb],
                /*c_mod=*/(short)0, acc, /*reuse_a=*/false, /*reuse_b=*/false);
        }
        // D 16x16 layout: VGPR r -> M = r + 8*half, N = m. Only n<12 exists.
        if (m < WIRES) {
            #pragma unroll
            for (int r = 0; r < 8; ++r) {
                const int mrow = r + 8 * half;
                const size_t grow = (size_t)b0 * WIRES + (size_t)(t * 16 + mrow);
                out[grow * WIRES + m] = acc[r] + bias;
            }
        }
    }
}

extern "C" void kernel_launch(void* const* d_in, const int* in_sizes, int n_in,
                              void* d_out, int out_size, void* d_ws, size_t ws_size,
                              hipStream_t stream) {
    const float* x     = (const float*)d_in[0];  // (B,12,12)
    const float* theta = (const float*)d_in[1];  // (12)
    const float* Wout  = (const float*)d_in[2];  // (12,12)
    const float* bout  = (const float*)d_in[3];  // (12)
    float* out = (float*)d_out;

    const int B = in_sizes[0] / (WIRES * WIRES); // 2048
    const int blocks = B / BATCH_PER_BLOCK;      // 128

    qsa_kernel<<<blocks, THREADS, 0, stream>>>(x, theta, Wout, bout, out);
}